// RNNModel_90159953477831
// MI455X (gfx1250) — compile-verified
//
#include <hip/hip_runtime.h>
#include <math.h>

#define NTOKEN 32000
#define NINP   512
#define NHID   2048
#define BB     8
#define SS     64
#define TT     64
#define CATW   (NINP + 3 * NHID)   // 6656
#define BIW    (2 * NHID)          // 4096

typedef __attribute__((ext_vector_type(2))) float v2f;
typedef __attribute__((ext_vector_type(8))) float v8f;

// ---------------------------------------------------------------------------
// fp32 WMMA GEMM body:  D[M,N] = act( A[M,K] @ W[N,K]^T + addend + bias )
// Block = 256 threads = 8 waves, split as (8/SK) column-groups x SK K-slices.
// Each wave computes a 16x32 output strip over K/SK; for SK>1 the partial
// accumulators are combined through LDS (one barrier), cutting the per-wave
// K critical path by SK (the sequential RNN/decode steps are latency-bound
// on this chain, not on bandwidth or wave count).
// The K loop is branchless with full EXEC (WMMA requirement): out-of-range
// M rows / N cols use clamped duplicate loads whose outputs the epilogue
// discards.  Manual 1-iteration software pipelining keeps loads ahead of
// the WMMAs independent of the register allocator.
// act: 0=none, 1=tanh, 2=sigmoid
// ---------------------------------------------------------------------------
template <int SK>
__device__ __forceinline__ void gemm_body(
    const float* __restrict__ A, const float* __restrict__ W,
    const float* __restrict__ bias,
    const float* __restrict__ addend, long long add_rs, long long add_off,
    float* __restrict__ D,
    float* __restrict__ D2, long long d2_rs, long long d2_off,
    int M, int N, int K, int act, int nblk) {
  constexpr int CG = 8 / SK;                 // column-groups per block
  const int lane = threadIdx.x & 31;
  const int wave = threadIdx.x >> 5;
  const int cg   = wave % CG;
  const int ks   = wave / CG;
  const int m0   = blockIdx.x * 16;
  const int n0   = (nblk * CG + cg) * 32;
  if (SK == 1 && n0 >= N) return;            // only legal when no barrier

  // fp32 16x16x4 fragment: lanes 0-15 -> K pair {0,1}, lanes 16-31 -> {2,3}
  const int kk_of = (lane >> 4) << 1;
  const int arow  = min(m0 + (lane & 15), M - 1);   // clamped duplicate
  const int nc0   = min(n0 + (lane & 15), N - 1);
  const int nc1   = min(n0 + 16 + (lane & 15), N - 1);

  const float* Ap = A + (long long)arow * K + kk_of;
  const float* W0 = W + (long long)nc0 * K + kk_of;
  const float* W1 = W + (long long)nc1 * K + kk_of;

  const int kbeg = ks * (K / SK);
  const int kend = kbeg + (K / SK);

  v8f acc0 = {};
  v8f acc1 = {};
  int k = kbeg;
  v2f a  = *(const v2f*)(Ap + k);
  v2f b0 = *(const v2f*)(W0 + k);
  v2f b1 = *(const v2f*)(W1 + k);
#pragma unroll 4
  for (k = kbeg + 4; k < kend; k += 4) {
    const v2f an  = *(const v2f*)(Ap + k);   // issue next-iter loads first
    const v2f b0n = *(const v2f*)(W0 + k);
    const v2f b1n = *(const v2f*)(W1 + k);
    acc0 = __builtin_amdgcn_wmma_f32_16x16x4_f32(false, a, false, b0,
                                                 (short)0, acc0, false, false);
    acc1 = __builtin_amdgcn_wmma_f32_16x16x4_f32(false, a, false, b1,
                                                 (short)0, acc1, false, false);
    a = an; b0 = b0n; b1 = b1n;
  }
  acc0 = __builtin_amdgcn_wmma_f32_16x16x4_f32(false, a, false, b0,
                                               (short)0, acc0, false, false);
  acc1 = __builtin_amdgcn_wmma_f32_16x16x4_f32(false, a, false, b1,
                                               (short)0, acc1, false, false);

  // ---- split-K reduction through LDS ----
  if (SK > 1) {
    __shared__ float lds[(SK > 1 ? (SK - 1) * CG * 512 : 1)];
    if (ks > 0) {
      const int base = ((ks - 1) * CG + cg) * 512 + lane * 16;
#pragma unroll
      for (int i = 0; i < 8; ++i) {
        lds[base + i]     = acc0[i];
        lds[base + 8 + i] = acc1[i];
      }
    }
    __syncthreads();
    if (ks == 0) {
#pragma unroll
      for (int j = 1; j < SK; ++j) {
        const int base = ((j - 1) * CG + cg) * 512 + lane * 16;
#pragma unroll
        for (int i = 0; i < 8; ++i) {
          acc0[i] += lds[base + i];
          acc1[i] += lds[base + 8 + i];
        }
      }
    }
  }
  if (ks != 0) return;

  // C/D layout: VGPR r -> (lanes 0-15: M=m0+r, N=lane), (lanes 16-31: M=m0+r+8)
  const int colL  = lane & 15;
  const int rbase = (lane >> 4) << 3;
#pragma unroll
  for (int r = 0; r < 8; ++r) {
    const int row = m0 + rbase + r;
    if (row >= M) continue;
#pragma unroll
    for (int half = 0; half < 2; ++half) {
      const int col = n0 + half * 16 + colL;
      if (col >= N) continue;
      float v = half ? acc1[r] : acc0[r];
      if (bias)   v += bias[col];
      if (addend) v += addend[(long long)row * add_rs + add_off + col];
      if (act == 1)      v = tanhf(v);
      else if (act == 2) v = 1.0f / (1.0f + __expf(-v));
      D[(long long)row * N + col] = v;
      if (D2) D2[(long long)row * d2_rs + d2_off + col] = v;
    }
  }
}

template <int SK>
__global__ void gemm_wmma_nt(const float* __restrict__ A,
                             const float* __restrict__ W,
                             const float* __restrict__ bias,
                             const float* __restrict__ addend,
                             long long add_rs, long long add_off,
                             float* __restrict__ D,
                             float* __restrict__ D2,
                             long long d2_rs, long long d2_off,
                             int M, int N, int K, int act) {
  gemm_body<SK>(A, W, bias, addend, add_rs, add_off,
                D, D2, d2_rs, d2_off, M, N, K, act, blockIdx.y);
}

// ---------------------------------------------------------------------------
// Fused fwd+bwd Elman step (blockIdx.z = direction), one launch per t:
//   h_new = tanh(pre[:,pos,:] + h @ w_hh^T + b_hh); scatter into bi_src.
// ---------------------------------------------------------------------------
__global__ void rnn_step_dual(const float* __restrict__ hprev_f,
                              const float* __restrict__ hprev_b,
                              const float* __restrict__ whh_f,
                              const float* __restrict__ whh_b,
                              const float* __restrict__ bhh_f,
                              const float* __restrict__ bhh_b,
                              const float* __restrict__ pre_f,
                              const float* __restrict__ pre_b,
                              float* __restrict__ hnew_f,
                              float* __restrict__ hnew_b,
                              float* __restrict__ bi, int t) {
  const int dir = blockIdx.z;
  const int pos = dir ? (SS - 1 - t) : t;
  const float* hp  = dir ? hprev_b : hprev_f;
  const float* whh = dir ? whh_b : whh_f;
  const float* bhh = dir ? bhh_b : bhh_f;
  const float* pre = dir ? pre_b : pre_f;
  float* hn = dir ? hnew_b : hnew_f;
  gemm_body<4>(hp, whh, bhh,
               pre, (long long)SS * NHID, (long long)pos * NHID,
               hn,
               bi, (long long)SS * BIW, (long long)pos * BIW + dir * NHID,
               BB, NHID, NHID, 1, blockIdx.y);
}

// ---------------------------------------------------------------------------
// Embedding gather: rows 0..B*S-1 from src table, rest from tgt table.
// ---------------------------------------------------------------------------
__global__ void gather_embed(const int* __restrict__ src,
                             const int* __restrict__ tgt,
                             const float* __restrict__ emb,
                             float* __restrict__ emb_src,
                             float* __restrict__ emb_tgt) {
  const int r = blockIdx.x;
  const int nsrc = BB * SS;
  int tok;
  float* dst;
  if (r < nsrc) { tok = src[r];        dst = emb_src + (long long)r * NINP; }
  else          { tok = tgt[r - nsrc]; dst = emb_tgt + (long long)(r - nsrc) * NINP; }
  const float* e = emb + (long long)tok * NINP;
  for (int k = threadIdx.x; k < NINP; k += 256) dst[k] = e[k];
}

__global__ void zero_fill(float* __restrict__ p, long long n) {
  long long i = (long long)blockIdx.x * blockDim.x + threadIdx.x;
  if (i < n) p[i] = 0.0f;
}

// ---------------------------------------------------------------------------
// Attention precompute: score_h[b,s] = bi[b,s,:]·wh_w + wh_b ; mask from src==0
// ---------------------------------------------------------------------------
__global__ void attn_pre(const float* __restrict__ bi,
                         const float* __restrict__ wh_w,
                         const float* __restrict__ wh_b,
                         const int* __restrict__ src,
                         float* __restrict__ score_h,
                         float* __restrict__ mask) {
  const int rs = blockIdx.x;
  const float* row = bi + (long long)rs * BIW;
  __shared__ float red[256];
  float s = 0.0f;
  for (int k = threadIdx.x; k < BIW; k += 256) s += row[k] * wh_w[k];
  red[threadIdx.x] = s;
  __syncthreads();
  for (int st = 128; st > 0; st >>= 1) {
    if (threadIdx.x < st) red[threadIdx.x] += red[threadIdx.x + st];
    __syncthreads();
  }
  if (threadIdx.x == 0) {
    score_h[rs] = red[0] + wh_b[0];
    mask[rs]    = (src[rs] == 0) ? -999999999.0f : 0.0f;
  }
}

// ---------------------------------------------------------------------------
// One decode attention step, block per batch b:
//   s_b = h·ws_w + ws_b; att = softmax(sigmoid(s_b + score_h) + mask)
//   c   = att @ bi_src ;  cat[b] = [emb_tgt[b,t], h[b], c]
// ---------------------------------------------------------------------------
__global__ void attn_step(const float* __restrict__ h,
                          const float* __restrict__ emb_tgt,
                          const float* __restrict__ bi,
                          const float* __restrict__ ws_w,
                          const float* __restrict__ ws_b,
                          const float* __restrict__ score_h,
                          const float* __restrict__ mask,
                          float* __restrict__ cat, int t) {
  const int b = blockIdx.x;
  __shared__ float red[256];
  __shared__ float att[SS];
  const float* hb = h + (long long)b * NHID;

  float s = 0.0f;
  for (int k = threadIdx.x; k < NHID; k += 256) s += hb[k] * ws_w[k];
  red[threadIdx.x] = s;
  __syncthreads();
  for (int st = 128; st > 0; st >>= 1) {
    if (threadIdx.x < st) red[threadIdx.x] += red[threadIdx.x + st];
    __syncthreads();
  }
  const float sb = red[0] + ws_b[0];
  __syncthreads();

  if (threadIdx.x < SS) {
    const float z = sb + score_h[b * SS + threadIdx.x];
    att[threadIdx.x] = 1.0f / (1.0f + __expf(-z)) + mask[b * SS + threadIdx.x];
  }
  __syncthreads();
  if (threadIdx.x == 0) {  // softmax over only 64 entries: serial is cheap
    float mx = -1e30f;
    for (int i = 0; i < SS; ++i) mx = fmaxf(mx, att[i]);
    float sum = 0.0f;
    for (int i = 0; i < SS; ++i) { const float e = __expf(att[i] - mx); att[i] = e; sum += e; }
    const float inv = 1.0f / sum;
    for (int i = 0; i < SS; ++i) att[i] *= inv;
  }
  __syncthreads();

  float* cb = cat + (long long)b * CATW;
  for (int k = threadIdx.x; k < NINP; k += 256)
    cb[k] = emb_tgt[((long long)b * TT + t) * NINP + k];
  for (int k = threadIdx.x; k < NHID; k += 256)
    cb[NINP + k] = hb[k];
  for (int d = threadIdx.x; d < BIW; d += 256) {
    float c = 0.0f;
    const float* bib = bi + (long long)b * SS * BIW + d;
    for (int s2 = 0; s2 < SS; ++s2) c += att[s2] * bib[(long long)s2 * BIW];
    cb[NINP + NHID + d] = c;
  }
}

// ---------------------------------------------------------------------------
// In-place log_softmax over rows of 32000 (block per row).
// ---------------------------------------------------------------------------
__global__ void log_softmax_rows(float* __restrict__ out) {
  float* p = out + (long long)blockIdx.x * NTOKEN;
  __shared__ float red[256];
  float mx = -1e30f;
  for (int k = threadIdx.x; k < NTOKEN; k += 256) mx = fmaxf(mx, p[k]);
  red[threadIdx.x] = mx;
  __syncthreads();
  for (int st = 128; st > 0; st >>= 1) {
    if (threadIdx.x < st) red[threadIdx.x] = fmaxf(red[threadIdx.x], red[threadIdx.x + st]);
    __syncthreads();
  }
  const float M = red[0];
  __syncthreads();
  float s = 0.0f;
  for (int k = threadIdx.x; k < NTOKEN; k += 256) s += __expf(p[k] - M);
  red[threadIdx.x] = s;
  __syncthreads();
  for (int st = 128; st > 0; st >>= 1) {
    if (threadIdx.x < st) red[threadIdx.x] += red[threadIdx.x + st];
    __syncthreads();
  }
  const float lse = M + __logf(red[0]);
  for (int k = threadIdx.x; k < NTOKEN; k += 256) p[k] -= lse;
}

// ---------------------------------------------------------------------------
extern "C" void kernel_launch(void* const* d_in, const int* in_sizes, int n_in,
                              void* d_out, int out_size, void* d_ws, size_t ws_size,
                              hipStream_t stream) {
  (void)in_sizes; (void)n_in; (void)out_size; (void)ws_size;
  const int*   src    = (const int*)  d_in[0];
  const int*   tgt    = (const int*)  d_in[1];
  const float* emb    = (const float*)d_in[2];
  const float* w_ih_f = (const float*)d_in[3];
  const float* w_hh_f = (const float*)d_in[4];
  const float* b_ih_f = (const float*)d_in[5];
  const float* b_hh_f = (const float*)d_in[6];
  const float* w_ih_b = (const float*)d_in[7];
  const float* w_hh_b = (const float*)d_in[8];
  const float* b_ih_b = (const float*)d_in[9];
  const float* b_hh_b = (const float*)d_in[10];
  const float* ws_w   = (const float*)d_in[11];
  const float* ws_b   = (const float*)d_in[12];
  const float* wh_w   = (const float*)d_in[13];
  const float* wh_b   = (const float*)d_in[14];
  const float* rnn_w  = (const float*)d_in[15];
  const float* rnn_b  = (const float*)d_in[16];
  const float* dec_w  = (const float*)d_in[17];
  const float* dec_b  = (const float*)d_in[18];
  float* out = (float*)d_out;

  // ---- workspace layout (floats) ----
  float* ws = (float*)d_ws;
  long long off = 0;
  float* emb_src = ws + off; off += (long long)BB * SS * NINP;
  float* emb_tgt = ws + off; off += (long long)BB * TT * NINP;
  float* pre_f   = ws + off; off += (long long)BB * SS * NHID;   // rows (b*S+s)
  float* pre_b   = ws + off; off += (long long)BB * SS * NHID;
  float* bi      = ws + off; off += (long long)BB * SS * BIW;    // [B,S,2H]
  float* hstate  = ws + off; off += 6LL * BB * NHID;             // 3 ping-pong pairs
  float* score_h = ws + off; off += (long long)BB * SS;
  float* maskb   = ws + off; off += (long long)BB * SS;
  float* cat     = ws + off; off += (long long)BB * CATW;
  float* out_hs  = ws + off; off += (long long)BB * TT * NHID;   // rows (b*T+t)

  float* hf[2] = { hstate + 0 * BB * NHID, hstate + 1 * BB * NHID };
  float* hb[2] = { hstate + 2 * BB * NHID, hstate + 3 * BB * NHID };
  float* hd[2] = { hstate + 4 * BB * NHID, hstate + 5 * BB * NHID };

  // 0) init recurrent states to zero every call (ws is poisoned by harness)
  {
    const long long n = 6LL * BB * NHID;
    zero_fill<<<(unsigned)((n + 255) / 256), 256, 0, stream>>>(hstate, n);
  }

  // 1) embedding gathers
  gather_embed<<<BB * SS + BB * TT, 256, 0, stream>>>(src, tgt, emb, emb_src, emb_tgt);

  // 2) input projections: pre = emb_src @ w_ih^T + b_ih   (M=512,N=2048,K=512)
  {
    dim3 g((BB * SS) / 16, NHID / 256);   // x = M tiles (fast), y = N groups
    gemm_wmma_nt<1><<<g, 256, 0, stream>>>(emb_src, w_ih_f, b_ih_f,
                                           nullptr, 0, 0, pre_f, nullptr, 0, 0,
                                           BB * SS, NHID, NINP, 0);
    gemm_wmma_nt<1><<<g, 256, 0, stream>>>(emb_src, w_ih_b, b_ih_b,
                                           nullptr, 0, 0, pre_b, nullptr, 0, 0,
                                           BB * SS, NHID, NINP, 0);
  }

  // 3) bidirectional Elman recurrence: 64 sequential fused launches.
  //    Split-K=4 shortens the per-wave critical path to K/4 = 512 elements;
  //    w_hh (16.8 MB each) stays L2-resident across all steps.
  for (int t = 0; t < SS; ++t) {
    dim3 g(1, NHID / 64, 2);              // 2 col-groups/block, z = direction
    rnn_step_dual<<<g, 256, 0, stream>>>(hf[t & 1], hb[t & 1],
                                         w_hh_f, w_hh_b, b_hh_f, b_hh_b,
                                         pre_f, pre_b,
                                         hf[(t + 1) & 1], hb[(t + 1) & 1],
                                         bi, t);
  }

  // 4) attention precompute: score_h, mask
  attn_pre<<<BB * SS, 256, 0, stream>>>(bi, wh_w, wh_b, src, score_h, maskb);

  // 5) decode loop: attention + context + concat, then cell GEMM (K=6656,
  //    split-K=4 -> per-wave chain 1664 elements). rnn_w (54.5 MB) L2-resident.
  for (int t = 0; t < TT; ++t) {
    attn_step<<<BB, 256, 0, stream>>>(hd[t & 1], emb_tgt, bi, ws_w, ws_b,
                                      score_h, maskb, cat, t);
    dim3 g(1, NHID / 64);
    gemm_wmma_nt<4><<<g, 256, 0, stream>>>(
        cat, rnn_w, rnn_b,
        nullptr, 0, 0,
        hd[(t + 1) & 1],
        out_hs, (long long)TT * NHID, (long long)t * NHID,
        BB, NHID, CATW, 2);
  }

  // 6) vocab decoder: logits = out_hs @ dec_w^T + dec_b  (M=512,N=32000,K=2048)
  //    M tiles vary fastest -> the 32 co-resident M-blocks of each 256-col
  //    dec_w slice (2 MB) share it through L2; dec_w crosses HBM ~once
  //    (262 MB @ 23.3 TB/s ~= 11 us floor), out_hs (4 MB) stays L2-resident.
  {
    dim3 g((BB * TT) / 16, NTOKEN / 256);
    gemm_wmma_nt<1><<<g, 256, 0, stream>>>(out_hs, dec_w, dec_b,
                                           nullptr, 0, 0, out, nullptr, 0, 0,
                                           BB * TT, NTOKEN, NHID, 0);
  }

  // 7) in-place log_softmax over each vocab row
  log_softmax_rows<<<BB * TT, 256, 0, stream>>>(out);
}